// CorrBlockSingleScale_31318901522816
// MI455X (gfx1250) — compile-verified
//
#include <hip/hip_runtime.h>
#include <hip/hip_bf16.h>

// ---------------------------------------------------------------------------
// RAFT CorrBlock single-scale lookup, sparse formulation for MI455X (gfx1250).
//
//   B=1, C=256, H=64, W=96, R=4  ->  HW = 6144 queries, 81 output channels.
//
// Each wave computes only the 10x10 integer tap window its query samples
// (100 length-256 dot products, 1.6% of the dense corr-volume FLOPs) with
// V_WMMA_F32_16X16X4_F32. Invalid/OOB taps gather from a clamped safe
// address and are zeroed at the store, keeping the K-chain fully branchless
// (EXEC stays all-ones across the WMMAs, per ISA 7.12 requirement).
// ---------------------------------------------------------------------------

typedef __attribute__((ext_vector_type(2))) float v2f;
typedef __attribute__((ext_vector_type(8))) float v8f;

namespace {
constexpr int C   = 256;
constexpr int H   = 64;
constexpr int W   = 96;
constexpr int R   = 4;
constexpr int HW  = H * W;          // 6144 queries
constexpr int KK  = 2 * R + 1;      // 9
constexpr int NCH = KK * KK;        // 81 output channels
constexpr int TAPS = (KK + 1) * (KK + 1);   // 10x10 = 100 integer taps
constexpr int NTILE = (TAPS + 15) / 16;     // 7 WMMA N-tiles of 16 taps
constexpr int WAVES = 8;                    // waves per block (256 threads)
constexpr float INV_SQRT_C = 0.0625f;       // 1/sqrt(256)
}

__global__ __launch_bounds__(WAVES * 32)
void corr_lookup_wmma(const float* __restrict__ f1,
                      const float* __restrict__ f2,
                      const float* __restrict__ coords,
                      float* __restrict__ out) {
  __shared__ float lds_f1[WAVES][C];          // query feature vector per wave
  __shared__ float lds_T[WAVES][10 * 10 + 8]; // 10x10 tap correlations

  const int lane = threadIdx.x & 31;
  const int wid  = threadIdx.x >> 5;
  const int q    = blockIdx.x * WAVES + wid;   // grid sized so q < HW always

  // ---- stage this query's f1 column (stride HW, channel-major) into LDS ----
  #pragma unroll
  for (int i = 0; i < C / 32; ++i) {
    const int c = i * 32 + lane;
    lds_f1[wid][c] = f1[(size_t)c * HW + q];
  }

  // ---- centroid: coords[0]=x, coords[1]=y ----
  const float cx = coords[q];
  const float cy = coords[HW + q];
  const float xbf = floorf(cx);
  const float ybf = floorf(cy);
  const int   xb  = (int)xbf;
  const int   yb  = (int)ybf;
  const float fx  = cx - xbf;
  const float fy  = cy - ybf;

  __syncthreads();   // lds_f1 visible to all lanes

  // ---- WMMA f32 16x16x4 operand layout (per ISA 7.12.2, 32-bit 16x4):
  //      lanes 0-15 carry K = k0+0,k0+1 ; lanes 16-31 carry K = k0+2,k0+3.
  const int khalf = (lane >> 4) * 2;
  const int nlane = lane & 15;

  for (int tile = 0; tile < NTILE; ++tile) {
    const int t  = tile * 16 + nlane;          // tap index 0..111
    const int m  = t / 10;                     // tap row    (y direction)
    const int n  = t - m * 10;                 // tap column (x direction)
    const int yi = yb - R + m;
    const int xi = xb - R + n;
    const bool valid = (t < TAPS) & (xi >= 0) & (xi < W) & (yi >= 0) & (yi < H);
    // Clamp invalid taps to a safe in-bounds column; their (finite) garbage
    // contributions live only in their own D column and are discarded below.
    const float* __restrict__ bcol = f2 + (valid ? (yi * W + xi) : 0);

    v8f acc = {};
    #pragma unroll 8
    for (int k0 = 0; k0 < C; k0 += 4) {
      const int kb = k0 + khalf;
      v2f a, b;
      a.x = lds_f1[wid][kb];
      a.y = lds_f1[wid][kb + 1];
      b.x = bcol[(size_t)kb * HW];            // unconditional, branchless
      b.y = bcol[(size_t)(kb + 1) * HW];
      // D = A(16x4) * B(4x16) + C ; all 16 M-rows identical (A broadcast).
      acc = __builtin_amdgcn_wmma_f32_16x16x4_f32(
          /*neg_a=*/false, a, /*neg_b=*/false, b,
          /*c_mod=*/(short)0, acc, /*reuse_a=*/false, /*reuse_b=*/false);
    }
    // D row M=0 lives in VGPR0 of lanes 0-15 (N = lane).
    const float tv = valid ? acc[0] * INV_SQRT_C : 0.0f;
    if (lane < 16 && t < TAPS) lds_T[wid][m * 10 + n] = tv;
  }

  __syncthreads();   // lds_T visible to all lanes

  // ---- bilinear combine. Reference meshgrid 'ij' quirk:
  //      channel ch = i*9 + j samples at (x = cx + (i-4), y = cy + (j-4)).
  const float w00 = (1.0f - fy) * (1.0f - fx);
  const float w01 = (1.0f - fy) * fx;
  const float w10 = fy * (1.0f - fx);
  const float w11 = fy * fx;

  for (int ch = lane; ch < NCH; ch += 32) {
    const int ix = ch / KK;        // x-offset index -> tap column ix, ix+1
    const int jy = ch - ix * KK;   // y-offset index -> tap row    jy, jy+1
    const float t00 = lds_T[wid][jy * 10 + ix];
    const float t01 = lds_T[wid][jy * 10 + ix + 1];
    const float t10 = lds_T[wid][(jy + 1) * 10 + ix];
    const float t11 = lds_T[wid][(jy + 1) * 10 + ix + 1];
    out[(size_t)ch * HW + q] = w00 * t00 + w01 * t01 + w10 * t10 + w11 * t11;
  }
}

extern "C" void kernel_launch(void* const* d_in, const int* in_sizes, int n_in,
                              void* d_out, int out_size, void* d_ws, size_t ws_size,
                              hipStream_t stream) {
  (void)in_sizes; (void)n_in; (void)out_size; (void)d_ws; (void)ws_size;
  const float* f1     = (const float*)d_in[0];
  const float* f2     = (const float*)d_in[1];
  const float* coords = (const float*)d_in[2];
  float* out          = (float*)d_out;

  const int blocks = HW / WAVES;   // 768 blocks x 8 waves = 6144 queries
  hipLaunchKernelGGL(corr_lookup_wmma, dim3(blocks), dim3(WAVES * 32), 0, stream,
                     f1, f2, coords, out);
}